// RelationModel_32538672235025
// MI455X (gfx1250) — compile-verified
//
#include <hip/hip_runtime.h>

// ---------------------------------------------------------------------------
// RelationModel head for MI455X (gfx1250, wave32, WMMA).
//
//   x = concat(hidden[b,0], hidden[b,e1], hidden[b,e2])   [P, 3072]
//   h = relu(x @ W1 + b1)                                 [P, 512]
//   logits = h @ W2 + b2                                  [P, 13]
//
// fc1 = 206 GFLOP fp32 -> V_WMMA_F32_16X16X4_F32 (exact fp32 accumulate).
// hidden (64MB) + W1 (6MB) are L2-resident (192MB L2): the kernel is an
// L2->LDS staging + matrix-core pipeline, not an HBM streamer.
//
// Pipelining: next chunk's global loads are issued into registers right after
// the top barrier, the current chunk is computed from LDS, and the registers
// are ds_store'd just before the next barrier -> L2 latency hidden behind
// 64 WMMAs of compute, with a single ~62KB LDS buffer.
// ---------------------------------------------------------------------------

typedef __attribute__((ext_vector_type(2))) float v2f;
typedef __attribute__((ext_vector_type(8))) float v8f;

#define S_   512
#define H_   1024
#define H3   3072      // 3*H
#define N1   512       // H/2
#define P_   65536
#define R_   13

#define ROWS      256          // pairs per workgroup
#define COLS      128          // fc1 output columns per workgroup
#define KC        32           // K-chunk staged per iteration
#define NTHREADS  512          // 16 wave32s
#define A_STRIDE  36           // KC + 4 pad  -> conflict-free ds_load_b64 of A
#define BP_STRIDE 288          // 2*COLS + 32 pad, pair-interleaved W1 rows
#define NCHUNK    (H3 / KC)    // 96

__global__ __launch_bounds__(NTHREADS)
void relation_fc_kernel(const float* __restrict__ hidden,
                        const float* __restrict__ W1,
                        const float* __restrict__ b1,
                        const float* __restrict__ W2,
                        const int*   __restrict__ pb,
                        const int*   __restrict__ e1,
                        const int*   __restrict__ e2,
                        float*       __restrict__ out)
{
    // ~62.5 KB static LDS total (safe under CU-mode caps)
    __shared__ __align__(16) float sA[ROWS * A_STRIDE];        // 36864 B
    __shared__ __align__(16) float sB[(KC / 2) * BP_STRIDE];   // 18432 B
    __shared__ __align__(16) float sW2[COLS * R_];             //  6656 B
    __shared__ __align__(16) float sB1[COLS];                  //   512 B

    const int tid     = threadIdx.x;
    const int rowBase = blockIdx.x * ROWS;
    const int colBase = blockIdx.y * COLS;

    // Loader role: 2 threads per pair-row, each stages 16 contiguous floats.
    const int lrow  = tid >> 1;           // 0..255
    const int lpart = tid & 1;            // which 16-float half of the chunk
    const int p   = rowBase + lrow;
    const int pbv = pb[p];
    const float* seg0 = hidden + ((size_t)pbv * S_ + 0)      * H_;
    const float* seg1 = hidden + ((size_t)pbv * S_ + e1[p])  * H_;
    const float* seg2 = hidden + ((size_t)pbv * S_ + e2[p])  * H_;

    // Compute role: wave w owns rows [w*16, w*16+16) x all 128 cols (8 tiles).
    const int wave = tid >> 5;            // 0..15
    const int lane = tid & 31;
    const int nl   = lane & 15;           // N (and M for A operand) in-tile idx
    const int hh   = lane >> 4;           // half-wave selects K pair
    const int m0   = wave * 16;

    // Register staging for the software pipeline (global -> regs -> LDS).
    float4 ar[4];                         // 16 floats of one gathered A row
    float4 br[2];                         // 2 float4 of W1

    auto issue_loads = [&](int kb) {
        const int seg = kb >> 10;                  // 0: cls, 1: e1, 2: e2
        const float* srow = (seg == 0) ? seg0 : (seg == 1) ? seg1 : seg2;
        const float* src  = srow + (kb & 1023) + lpart * 16;
#pragma unroll
        for (int i = 0; i < 4; ++i)
            ar[i] = *(const float4*)(src + i * 4);
#pragma unroll
        for (int j = 0; j < 2; ++j) {
            const int q  = tid * 2 + j;            // 0..1023 float4s
            const int k  = q >> 5;                 // 0..31
            const int c4 = (q & 31) * 4;           // col group
            br[j] = *(const float4*)&W1[(size_t)(kb + k) * N1 + colBase + c4];
        }
    };

    auto store_stage = [&]() {
        float* dst = &sA[lrow * A_STRIDE + lpart * 16];
#pragma unroll
        for (int i = 0; i < 4; ++i)
            *(float4*)(dst + i * 4) = ar[i];
        // pair-interleaved W1 layout: (k,n) -> sB[(k>>1)*BP_STRIDE + 2n + (k&1)]
        // so a WMMA B operand (rows k,k+1 at col n) is a single ds_load_b64.
#pragma unroll
        for (int j = 0; j < 2; ++j) {
            const int q  = tid * 2 + j;
            const int k  = q >> 5;
            const int c4 = (q & 31) * 4;
            float* d = &sB[(k >> 1) * BP_STRIDE + (k & 1)];
            d[(c4 + 0) * 2] = br[j].x;
            d[(c4 + 1) * 2] = br[j].y;
            d[(c4 + 2) * 2] = br[j].z;
            d[(c4 + 3) * 2] = br[j].w;
        }
    };

    v8f c[8];
    const v8f vzero = {0.f, 0.f, 0.f, 0.f, 0.f, 0.f, 0.f, 0.f};
#pragma unroll
    for (int t = 0; t < 8; ++t) c[t] = vzero;

    // ---- software-pipelined fc1 main loop ----
    issue_loads(0);
    store_stage();

    for (int ch = 0; ch < NCHUNK; ++ch) {
        __syncthreads();                           // stage visible to all
        const bool more = (ch + 1) < NCHUNK;
        if (more) issue_loads((ch + 1) * KC);      // L2 loads fly over compute

        // 8 K-steps x 8 N-tiles of V_WMMA_F32_16X16X4_F32.
        // A operand layout (ISA 7.12.2): lane = M + 16*(K/2), vgpr v = K%2.
#pragma unroll
        for (int ks = 0; ks < KC / 4; ++ks) {
            const int k0 = ks * 4;
            const v2f a = *(const v2f*)&sA[(m0 + nl) * A_STRIDE + k0 + 2 * hh];
            v2f bv[8];
#pragma unroll
            for (int t = 0; t < 8; ++t)
                bv[t] = *(const v2f*)&sB[((k0 >> 1) + hh) * BP_STRIDE +
                                         (t * 16 + nl) * 2];
#pragma unroll
            for (int t = 0; t < 8; ++t)
                c[t] = __builtin_amdgcn_wmma_f32_16x16x4_f32(
                    /*neg_a=*/false, a, /*neg_b=*/false, bv[t],
                    /*c_mod=*/(short)0, c[t],
                    /*reuse_a=*/false, /*reuse_b=*/false);
        }

        __syncthreads();                           // all reads of stage done
        if (more) store_stage();                   // regs -> LDS for next chunk
    }

    // ---- stage b1 / W2 slices for this column block ----
    __syncthreads();
    for (int i = tid; i < COLS * R_; i += NTHREADS)
        sW2[i] = W2[(size_t)(colBase + i / R_) * R_ + (i % R_)];
    for (int i = tid; i < COLS; i += NTHREADS)
        sB1[i] = b1[colBase + i];
    __syncthreads();

    // ---- bias + ReLU in registers (C/D layout: vgpr r -> M=r+8*hh, N=nl) ----
#pragma unroll
    for (int t = 0; t < 8; ++t) {
        const float bias = sB1[t * 16 + nl];
#pragma unroll
        for (int r = 0; r < 8; ++r) {
            const float v = c[t][r] + bias;
            c[t][r] = v > 0.f ? v : 0.f;
        }
    }

    // ---- fc2 partial over this 128-col block, lane-tree reduce, atomic ----
    for (int j = 0; j < R_; ++j) {
        float w[8];
#pragma unroll
        for (int t = 0; t < 8; ++t) w[t] = sW2[(t * 16 + nl) * R_ + j];
#pragma unroll
        for (int r = 0; r < 8; ++r) {
            float acc = 0.f;
#pragma unroll
            for (int t = 0; t < 8; ++t) acc += c[t][r] * w[t];
            // reduce across the 16 N-lanes of each half-wave
#pragma unroll
            for (int off = 1; off < 16; off <<= 1)
                acc += __shfl_xor(acc, off, 32);
            if (nl == 0) {
                const int row = rowBase + m0 + r + 8 * hh;
#if defined(__HIP_DEVICE_COMPILE__)
                unsafeAtomicAdd(&out[(size_t)row * R_ + j], acc);
#else
                atomicAdd(&out[(size_t)row * R_ + j], acc);
#endif
            }
        }
    }
}

// Pre-fill logits with b2 so the GEMM kernel can accumulate column-block
// partials with global fp32 atomics.
__global__ void init_out_kernel(const float* __restrict__ b2,
                                float* __restrict__ out)
{
    const int i = blockIdx.x * blockDim.x + threadIdx.x;
    if (i < P_ * R_) out[i] = b2[i % R_];
}

extern "C" void kernel_launch(void* const* d_in, const int* in_sizes, int n_in,
                              void* d_out, int out_size, void* d_ws, size_t ws_size,
                              hipStream_t stream)
{
    (void)in_sizes; (void)n_in; (void)out_size; (void)d_ws; (void)ws_size;
    const float* hidden = (const float*)d_in[0];
    const float* W1     = (const float*)d_in[1];
    const float* b1     = (const float*)d_in[2];
    const float* W2     = (const float*)d_in[3];
    const float* b2     = (const float*)d_in[4];
    const int*   pb     = (const int*)d_in[5];
    const int*   e1     = (const int*)d_in[6];
    const int*   e2     = (const int*)d_in[7];
    float* out = (float*)d_out;

    init_out_kernel<<<(P_ * R_ + 255) / 256, 256, 0, stream>>>(b2, out);

    dim3 grid(P_ / ROWS, N1 / COLS);   // 256 x 4 workgroups
    relation_fc_kernel<<<grid, NTHREADS, 0, stream>>>(hidden, W1, b1, W2,
                                                      pb, e1, e2, out);
}